// SparseGraphEncoder_16578573762731
// MI455X (gfx1250) — compile-verified
//
#include <hip/hip_runtime.h>
#include <hip/hip_bf16.h>

#define N_NODES   50000
#define N_EDGES   300000
#define HIDDEN    256
#define Z_DIM     128
#define N_LAYERS  3
#define NUM_GRAPHS 16
#define EPS       1e-5f

typedef float v2f __attribute__((ext_vector_type(2)));
typedef float v8f __attribute__((ext_vector_type(8)));

// ---------------------------------------------------------------- utilities

__global__ void zero_kernel(float* __restrict__ p, int n) {
    int i = blockIdx.x * blockDim.x + threadIdx.x;
    if (i < n) p[i] = 0.0f;
}

// blockDim.x == 256 assumed
__device__ __forceinline__ float block_reduce_sum(float v, float* red, int c) {
    red[c] = v;
    __syncthreads();
    for (int s = 128; s > 0; s >>= 1) {
        if (c < s) red[c] += red[c + s];
        __syncthreads();
    }
    float r = red[0];
    __syncthreads();
    return r;
}

// ---------------------------------------------------------------- degree / norm

__global__ void deg_accum_kernel(const int* __restrict__ dst, float* __restrict__ deg, int ne) {
    int e = blockIdx.x * blockDim.x + threadIdx.x;
    if (e < ne) atomicAdd(&deg[dst[e]], 1.0f);
}

// deg -> dinv in place; +1 accounts for the self loop, so deg >= 1 always
__global__ void dinv_kernel(float* __restrict__ deg, int n) {
    int i = blockIdx.x * blockDim.x + threadIdx.x;
    if (i < n) deg[i] = rsqrtf(deg[i] + 1.0f);
}

// ---------------------------------------------------------------- input projection

// one block (256 threads) per node: Linear(3,256) -> ReLU -> GroupNorm(1,256)
__global__ __launch_bounds__(256) void input_proj_kernel(
    const float* __restrict__ pos, const float* __restrict__ Win,
    const float* __restrict__ bin, const float* __restrict__ g,
    const float* __restrict__ be, float* __restrict__ X) {
    __shared__ float red[256];
    const int i = blockIdx.x, c = threadIdx.x;
    const float p0 = pos[3 * i + 0], p1 = pos[3 * i + 1], p2 = pos[3 * i + 2];
    float v = p0 * Win[0 * HIDDEN + c] + p1 * Win[1 * HIDDEN + c] + p2 * Win[2 * HIDDEN + c] + bin[c];
    v = fmaxf(v, 0.0f);
    const float mu  = block_reduce_sum(v, red, c) * (1.0f / HIDDEN);
    const float dlt = v - mu;
    const float var = block_reduce_sum(dlt * dlt, red, c) * (1.0f / HIDDEN);
    X[(size_t)i * HIDDEN + c] = dlt * rsqrtf(var + EPS) * g[c] + be[c];
}

// ---------------------------------------------------------------- GCN GEMM (f32 WMMA)

// H = X @ W and ACC = H * dinv(row)^2 (self-loop init, fused from registers).
// X: (50000,256) row-major, W: (256,256) row-major.
// Block = 16 waves (512 threads); wave ty owns N-tile ty, block owns M-tile blockIdx.x.
// The 16x256 A tile (16 KB) is staged once in LDS and shared by all 16 waves.
// 50000 == 3125*16, no M guard needed.
__global__ __launch_bounds__(512) void gcn_gemm_kernel(
    const float* __restrict__ X, const float* __restrict__ W,
    const float* __restrict__ dinv, float* __restrict__ H, float* __restrict__ ACC) {
    __shared__ float As[16 * HIDDEN];          // 16 KB A tile
    const int lane  = threadIdx.x;             // 0..31
    const int ntile = threadIdx.y;             // 0..15
    const int mtile = blockIdx.x;              // 0..3124
    const int tid   = ntile * 32 + lane;       // 0..511

    // cooperative A-tile load: 1024 float4s across 512 threads
    {
        const float4* Xv  = (const float4*)(X + (size_t)mtile * 16 * HIDDEN);
        float4*       Asv = (float4*)As;
#pragma unroll
        for (int idx = tid; idx < 16 * HIDDEN / 4; idx += 512) Asv[idx] = Xv[idx];
    }
    __syncthreads();

    const int m   = lane & 15;
    const int kh  = lane >> 4;                 // which K-half this lane holds
    const int col = ntile * 16 + (lane & 15);
    const float* ar = As + m * HIDDEN;

    v8f acc = {};
#pragma unroll 8
    for (int k = 0; k < HIDDEN; k += 4) {
        v2f a, b;
        // A 16x4: lanes 0-15 hold K={k,k+1}, lanes 16-31 hold K={k+2,k+3}  (ds_load_b64)
        a.x = ar[k + 2 * kh + 0];
        a.y = ar[k + 2 * kh + 1];
        // B 4x16 mirrored across lane halves (L2-resident W)
        b.x = W[(size_t)(k + 2 * kh + 0) * HIDDEN + col];
        b.y = W[(size_t)(k + 2 * kh + 1) * HIDDEN + col];
        acc = __builtin_amdgcn_wmma_f32_16x16x4_f32(false, a, false, b, (short)0, acc,
                                                    false, false);
    }
    // D 16x16 f32: VGPR v -> M = v + 8*(lane>=16), N = lane&15
    const int mbase = mtile * 16 + (lane >> 4) * 8;
#pragma unroll
    for (int v = 0; v < 8; ++v) {
        const int   row = mbase + v;
        const float di  = dinv[row];
        H[(size_t)row * HIDDEN + col]   = acc[v];
        ACC[(size_t)row * HIDDEN + col] = acc[v] * di * di;   // self-loop term
    }
}

// ---------------------------------------------------------------- message passing

// one wave per edge; lane covers 8 channels. acc/H are L2-resident (51 MB each).
__global__ __launch_bounds__(256) void scatter_kernel(
    const float* __restrict__ H, const float* __restrict__ dinv,
    const int* __restrict__ src, const int* __restrict__ dst, float* __restrict__ acc) {
    const int e = blockIdx.x * 8 + threadIdx.y;
    if (e >= N_EDGES) return;
    const int s = src[e], d = dst[e];
    const float w = dinv[s] * dinv[d];
    const float* hs = H + (size_t)s * HIDDEN;
    float* ad = acc + (size_t)d * HIDDEN;
#pragma unroll
    for (int c = threadIdx.x; c < HIDDEN; c += 32)
        atomicAdd(&ad[c], hs[c] * w);
}

// x = relu(groupnorm(acc + bias)) + x  (residual, in place)
// On the last layer also accumulates the per-graph mean-pool sums (fused).
__global__ __launch_bounds__(256) void post_layer_kernel(
    const float* __restrict__ acc, const float* __restrict__ bias,
    const float* __restrict__ g, const float* __restrict__ b, float* __restrict__ X,
    const int* __restrict__ batch, float* __restrict__ gsum, float* __restrict__ gcnt,
    int do_pool) {
    __shared__ float red[256];
    const int i = blockIdx.x, c = threadIdx.x;
    const float t   = acc[(size_t)i * HIDDEN + c] + bias[c];
    const float mu  = block_reduce_sum(t, red, c) * (1.0f / HIDDEN);
    const float dlt = t - mu;
    const float var = block_reduce_sum(dlt * dlt, red, c) * (1.0f / HIDDEN);
    const float y   = dlt * rsqrtf(var + EPS) * g[c] + b[c];
    const float xnew = fmaxf(y, 0.0f) + X[(size_t)i * HIDDEN + c];
    X[(size_t)i * HIDDEN + c] = xnew;
    if (do_pool) {
        const int gidx = batch[i];
        atomicAdd(&gsum[(size_t)gidx * HIDDEN + c], xnew);
        if (c == 0) atomicAdd(&gcnt[gidx], 1.0f);
    }
}

// ---------------------------------------------------------------- head

// single block, 256 threads: pooled MLP head -> out (16 x 128)
__global__ __launch_bounds__(256) void final_mlp_kernel(
    const float* __restrict__ gsum, const float* __restrict__ gcnt,
    const float* __restrict__ Wo1, const float* __restrict__ bo1,
    const float* __restrict__ go, const float* __restrict__ beo,
    const float* __restrict__ Wo2, const float* __restrict__ bo2,
    float* __restrict__ out) {
    __shared__ float pooled[NUM_GRAPHS * HIDDEN];
    __shared__ float t1[NUM_GRAPHS * HIDDEN];
    __shared__ float mu[NUM_GRAPHS], rs[NUM_GRAPHS];
    const int c = threadIdx.x;

    for (int gi = 0; gi < NUM_GRAPHS; ++gi) {
        const float cnt = fmaxf(gcnt[gi], 1.0f);
        pooled[gi * HIDDEN + c] = gsum[gi * HIDDEN + c] / cnt;
    }
    __syncthreads();

    for (int gi = 0; gi < NUM_GRAPHS; ++gi) {
        float s = bo1[c];
        for (int k = 0; k < HIDDEN; ++k) s += pooled[gi * HIDDEN + k] * Wo1[k * HIDDEN + c];
        t1[gi * HIDDEN + c] = fmaxf(s, 0.0f);
    }
    __syncthreads();

    if (c < NUM_GRAPHS) {
        float s = 0.0f;
        for (int k = 0; k < HIDDEN; ++k) s += t1[c * HIDDEN + k];
        const float m = s * (1.0f / HIDDEN);
        float v = 0.0f;
        for (int k = 0; k < HIDDEN; ++k) { float d = t1[c * HIDDEN + k] - m; v += d * d; }
        mu[c] = m;
        rs[c] = rsqrtf(v * (1.0f / HIDDEN) + EPS);
    }
    __syncthreads();

    for (int gi = 0; gi < NUM_GRAPHS; ++gi)
        t1[gi * HIDDEN + c] = (t1[gi * HIDDEN + c] - mu[gi]) * rs[gi] * go[c] + beo[c];
    __syncthreads();

    if (c < Z_DIM) {
        for (int gi = 0; gi < NUM_GRAPHS; ++gi) {
            float s = bo2[c];
            for (int k = 0; k < HIDDEN; ++k) s += t1[gi * HIDDEN + k] * Wo2[k * Z_DIM + c];
            out[gi * Z_DIM + c] = s;
        }
    }
}

// ---------------------------------------------------------------- launch

extern "C" void kernel_launch(void* const* d_in, const int* in_sizes, int n_in,
                              void* d_out, int out_size, void* d_ws, size_t ws_size,
                              hipStream_t stream) {
    const float* voxel_pos = (const float*)d_in[0];
    const int*   edge      = (const int*)d_in[1];   // (2, N_EDGES): [0]=src, [1]=dst
    const int*   batch     = (const int*)d_in[2];
    const float* W_in  = (const float*)d_in[3];
    const float* b_in  = (const float*)d_in[4];
    const float* g_in  = (const float*)d_in[5];
    const float* be_in = (const float*)d_in[6];
    const float* convW = (const float*)d_in[7];     // (3, 256, 256)
    const float* convB = (const float*)d_in[8];     // (3, 256)
    const float* gn_g  = (const float*)d_in[9];
    const float* gn_b  = (const float*)d_in[10];
    const float* W_o1  = (const float*)d_in[11];
    const float* b_o1  = (const float*)d_in[12];
    const float* g_o   = (const float*)d_in[13];
    const float* be_o  = (const float*)d_in[14];
    const float* W_o2  = (const float*)d_in[15];
    const float* b_o2  = (const float*)d_in[16];
    float* out = (float*)d_out;

    const int* e_src = edge;
    const int* e_dst = edge + N_EDGES;

    // workspace layout (floats): x | h | acc | dinv | gsum | gcnt  (~154 MB)
    float* ws   = (float*)d_ws;
    float* X    = ws;
    float* H    = X   + (size_t)N_NODES * HIDDEN;
    float* ACC  = H   + (size_t)N_NODES * HIDDEN;
    float* DINV = ACC + (size_t)N_NODES * HIDDEN;
    float* GSUM = DINV + N_NODES;
    float* GCNT = GSUM + NUM_GRAPHS * HIDDEN;

    // 1) zero accumulators (deg buffer + graph sums/counts, contiguous)
    zero_kernel<<<(N_NODES + 255) / 256, 256, 0, stream>>>(DINV, N_NODES);
    zero_kernel<<<(NUM_GRAPHS * HIDDEN + NUM_GRAPHS + 255) / 256, 256, 0, stream>>>(
        GSUM, NUM_GRAPHS * HIDDEN + NUM_GRAPHS);

    // 2) degrees over dst, then dinv = rsqrt(deg + 1) in place
    deg_accum_kernel<<<(N_EDGES + 255) / 256, 256, 0, stream>>>(e_dst, DINV, N_EDGES);
    dinv_kernel<<<(N_NODES + 255) / 256, 256, 0, stream>>>(DINV, N_NODES);

    // 3) input projection + groupnorm
    input_proj_kernel<<<N_NODES, 256, 0, stream>>>(voxel_pos, W_in, b_in, g_in, be_in, X);

    // 4) GCN layers (gemm fuses self-loop init; last post fuses pooling)
    for (int l = 0; l < N_LAYERS; ++l) {
        const float* Wl = convW + (size_t)l * HIDDEN * HIDDEN;
        gcn_gemm_kernel<<<N_NODES / 16, dim3(32, 16), 0, stream>>>(X, Wl, DINV, H, ACC);
        scatter_kernel<<<(N_EDGES + 7) / 8, dim3(32, 8), 0, stream>>>(H, DINV, e_src, e_dst, ACC);
        post_layer_kernel<<<N_NODES, 256, 0, stream>>>(
            ACC, convB + (size_t)l * HIDDEN, gn_g + (size_t)l * HIDDEN,
            gn_b + (size_t)l * HIDDEN, X, batch, GSUM, GCNT,
            (l == N_LAYERS - 1) ? 1 : 0);
    }

    // 5) head
    final_mlp_kernel<<<1, 256, 0, stream>>>(GSUM, GCNT, W_o1, b_o1, g_o, be_o, W_o2, b_o2, out);
}